// SpatialEncoding_56435870269640
// MI455X (gfx1250) — compile-verified
//
#include <hip/hip_runtime.h>

// SpatialEncoding scatter for MI455X (gfx1250, wave32).
// out[n,n] = 0; out[src[e], dst[e]] = b[min(path_len[e],5)-1], last duplicate wins.
// Pure HBM-bound scatter: NT b128 stores for zero-fill, u32 atomic-max for
// deterministic last-writer-wins, then a winner-check write pass.

#define MAX_PATH_DISTANCE 5
#define REF_D_FEAT 128

typedef float        v4f __attribute__((ext_vector_type(4)));
typedef unsigned int v4u __attribute__((ext_vector_type(4)));

// ---------------------------------------------------------------------------
// Pass 1: zero the 256MB output and the 256MB winner array with non-temporal
// 128-bit stores (streaming, write-once, larger than the 192MB L2).
// ---------------------------------------------------------------------------
__global__ void se_zero_kernel(float* __restrict__ out,
                               unsigned* __restrict__ winner,
                               long n_total) {
    const long tid    = (long)blockIdx.x * blockDim.x + threadIdx.x;
    const long stride = (long)gridDim.x * blockDim.x * 4;
    const v4f zf = {0.0f, 0.0f, 0.0f, 0.0f};
    const v4u zu = {0u, 0u, 0u, 0u};

    for (long p = tid * 4; p + 3 < n_total; p += stride) {
        __builtin_nontemporal_store(zf, (v4f*)(out + p));
        __builtin_nontemporal_store(zu, (v4u*)(winner + p));
    }
    // Scalar tail (n_total is 4-aligned for the reference shapes; safety only).
    if (tid == 0) {
        for (long p = n_total & ~3L; p < n_total; ++p) {
            out[p]    = 0.0f;
            winner[p] = 0u;
        }
    }
}

// ---------------------------------------------------------------------------
// Pass 2: deterministic last-writer-wins via atomic max on edge index (+1 so
// that "no writer" == 0). Lowers to global_atomic_max_u32 in the L2 atomic
// units.
// ---------------------------------------------------------------------------
__global__ void se_scatter_max_kernel(const int* __restrict__ src,
                                      const int* __restrict__ dst,
                                      unsigned* __restrict__ winner,
                                      long n_edges, long n_nodes) {
    const long stride = (long)gridDim.x * blockDim.x;
    for (long e = (long)blockIdx.x * blockDim.x + threadIdx.x;
         e < n_edges; e += stride) {
        const long s = (long)__builtin_nontemporal_load(src + e);
        const long d = (long)__builtin_nontemporal_load(dst + e);
        const long pos = s * n_nodes + d;
        atomicMax(&winner[pos], (unsigned)(e + 1));
    }
}

// ---------------------------------------------------------------------------
// Pass 3: the winning edge for each slot writes its bias value.
// ---------------------------------------------------------------------------
__global__ void se_scatter_write_kernel(const int* __restrict__ src,
                                        const int* __restrict__ dst,
                                        const int* __restrict__ path_len,
                                        const float* __restrict__ b,
                                        const unsigned* __restrict__ winner,
                                        float* __restrict__ out,
                                        long n_edges, long n_nodes) {
    const long stride = (long)gridDim.x * blockDim.x;
    for (long e = (long)blockIdx.x * blockDim.x + threadIdx.x;
         e < n_edges; e += stride) {
        const long s = (long)__builtin_nontemporal_load(src + e);
        const long d = (long)__builtin_nontemporal_load(dst + e);
        const long pos = s * n_nodes + d;
        if (winner[pos] == (unsigned)(e + 1)) {
            int len = __builtin_nontemporal_load(path_len + e);
            int idx = (len < MAX_PATH_DISTANCE ? len : MAX_PATH_DISTANCE) - 1;
            out[pos] = b[idx];
        }
    }
}

extern "C" void kernel_launch(void* const* d_in, const int* in_sizes, int n_in,
                              void* d_out, int out_size, void* d_ws, size_t ws_size,
                              hipStream_t stream) {
    // Inputs per reference setup_inputs() order:
    //   0: x        (N_NODES * D_FEAT) f32   -- only shape matters
    //   1: src      (N_PAIRS)          i32
    //   2: dst      (N_PAIRS)          i32
    //   3: path_len (N_PAIRS)          i32
    //   4: b        (MAX_PATH_DISTANCE) f32
    const int*   src  = (const int*)d_in[1];
    const int*   dst  = (const int*)d_in[2];
    const int*   plen = (const int*)d_in[3];
    const float* b    = (const float*)d_in[4];

    float*    out    = (float*)d_out;
    unsigned* winner = (unsigned*)d_ws;   // needs out_size * 4 bytes of scratch

    const long n_nodes = (long)in_sizes[0] / REF_D_FEAT;  // 8192
    const long n_edges = (long)in_sizes[1];               // 4,000,000
    const long n_total = (long)out_size;                  // n_nodes^2

    const int threads = 256;  // 8 wave32 waves per block

    // Pass 1: zero out + winner. 1M threads, each streams ~16 b128 NT stores
    // per array -> saturates the 23.3 TB/s HBM write path.
    {
        dim3 grid(4096);
        se_zero_kernel<<<grid, threads, 0, stream>>>(out, winner, n_total);
    }

    // Pass 2: atomic-max winner resolution (4M edges).
    {
        long blocks = (n_edges + threads - 1) / threads;
        if (blocks > 65535L * 16) blocks = 65535L * 16;
        se_scatter_max_kernel<<<dim3((unsigned)blocks), threads, 0, stream>>>(
            src, dst, winner, n_edges, n_nodes);
    }

    // Pass 3: winners write bias values.
    {
        long blocks = (n_edges + threads - 1) / threads;
        if (blocks > 65535L * 16) blocks = 65535L * 16;
        se_scatter_write_kernel<<<dim3((unsigned)blocks), threads, 0, stream>>>(
            src, dst, plen, b, winner, out, n_edges, n_nodes);
    }
}